// JointLoss_62036507623749
// MI455X (gfx1250) — compile-verified
//
#include <hip/hip_runtime.h>
#include <hip/hip_bf16.h>
#include <math.h>

// ---------------------------------------------------------------------------
// JointLoss for MI455X / gfx1250 (wave32, WMMA + async Global->LDS staging).
//   representation : [8192, 128] f32   (d_in[0])
//   xrecon         : [8192, 512] f32   (d_in[1])
//   xorig          : [8192, 512] f32   (d_in[2])
//   d_out : (loss, closs, recon_loss, zrecon_loss)
//
// sim = n @ n^T (8192x8192x128 fp32, 17.2 GFLOP) fused with streaming
// sum-exp (values bounded in [-2,2] -> no running max needed).
// Block = 8 waves sharing one column chunk: B tiles staged once per block
// into LDS (8x less L2 traffic), double-buffered via async-to-LDS DMA
// (ASYNCcnt) when available.  LDS layout is permuted so every wave-wide
// ds_load_b64 hits all 64 banks exactly once.
// ---------------------------------------------------------------------------

#define TWO_B     8192
#define BATCH     4096
#define KDIM      128
#define NCHUNK    4                   // column chunks per row sweep
#define CHUNK_COLS (TWO_B / NCHUNK)   // 2048
#define NTILES    (CHUNK_COLS / 16)   // 128
#define WAVES_PER_BLOCK 8
#define RECON_N   (TWO_B * 512)
#define ZREC_N    (BATCH * KDIM)

typedef float v2f __attribute__((ext_vector_type(2)));
typedef float v8f __attribute__((ext_vector_type(8)));

#if __has_builtin(__builtin_amdgcn_global_load_async_to_lds_b64)
#define HAVE_ASYNC_LDS 1
// pointee type per compiler diagnostic: int __attribute__((vector_size(8)))
typedef int async_b64_t __attribute__((vector_size(2 * sizeof(int))));
typedef async_b64_t __attribute__((address_space(1))) *async_gptr_t;  // global
typedef async_b64_t __attribute__((address_space(3))) *async_lptr_t;  // LDS
#else
#define HAVE_ASYNC_LDS 0
#endif

__device__ __forceinline__ void wait_async_le(int n) {
#if HAVE_ASYNC_LDS
#if __has_builtin(__builtin_amdgcn_s_wait_asynccnt)
  if (n == 0) __builtin_amdgcn_s_wait_asynccnt(0);
  else        __builtin_amdgcn_s_wait_asynccnt(4);
#else
  if (n == 0) asm volatile("s_wait_asynccnt 0x0" ::: "memory");
  else        asm volatile("s_wait_asynccnt 0x4" ::: "memory");
#endif
#else
  (void)n;   // fallback staging is synchronous ds_store; barrier suffices
#endif
}

__device__ __forceinline__ float wave_reduce_add(float v) {
  #pragma unroll
  for (int off = 16; off >= 1; off >>= 1) v += __shfl_xor(v, off, 32);
  return v;
}

// ---------------------------------------------------------------------------
__global__ void zero_acc_kernel(float* acc) {
  if (threadIdx.x < 4) acc[threadIdx.x] = 0.0f;
}

// ---------------------------------------------------------------------------
// row-normalize; fold sqrt(1/T)=sqrt(2) so WMMA output is sim/T directly.
// ---------------------------------------------------------------------------
__global__ __launch_bounds__(256)
void normalize_kernel(const float* __restrict__ rep, float* __restrict__ n) {
  const int wave = (blockIdx.x * blockDim.x + threadIdx.x) >> 5;  // row
  const int lane = threadIdx.x & 31;
  if (wave >= TWO_B) return;
  const float4 v = ((const float4*)(rep + (size_t)wave * KDIM))[lane];
  float ss = v.x * v.x + v.y * v.y + v.z * v.z + v.w * v.w;
  ss = wave_reduce_add(ss);
  const float scale = 1.41421356237309515f / (sqrtf(ss) + 1e-8f);
  float4 o;
  o.x = v.x * scale; o.y = v.y * scale; o.z = v.z * scale; o.w = v.w * scale;
  ((float4*)(n + (size_t)wave * KDIM))[lane] = o;
}

// ---------------------------------------------------------------------------
// Stage one 16-col B tile (16 x 128 f32 = 8KB) into an LDS buffer.
// Permuted layout: element e (0..127) of tile-col c lives at dword
//   (e>>2)*64 + 2*c + 32*((e>>1)&1) + (e&1)
// so reader lane (l,hi) at k-step loads dwords {2*laneid, 2*laneid+1} of
// slice k -> all 64 banks, conflict-free ds_load_b64.
// ---------------------------------------------------------------------------
__device__ __forceinline__ void stage_tile(const float* __restrict__ n,
                                           int cb, float* __restrict__ buf,
                                           int t) {
#if HAVE_ASYNC_LDS
  // global-coalesced: consecutive threads copy consecutive 8B pairs
  #pragma unroll
  for (int it = 0; it < 4; ++it) {
    const int q   = t + it * 256;     // pair index 0..1023
    const int col = q >> 6;           // tile column 0..15
    const int j   = q & 63;           // element-pair 0..63
    const float* gsrc = n + (size_t)(cb + col) * KDIM + 2 * j;
    float* ldst = buf + ((j >> 1) * 64 + 2 * col + 32 * (j & 1));
    __builtin_amdgcn_global_load_async_to_lds_b64(
        (async_gptr_t)gsrc, (async_lptr_t)ldst, 0, 0);
  }
#else
  // fallback: bank-friendly ds_store mapping (lane -> unique (col,parity))
  const int col = t & 15;
  const int par = (t >> 4) & 1;
  const int w   = t >> 5;
  #pragma unroll
  for (int it = 0; it < 4; ++it) {
    const int a = w * 4 + it;         // 0..31
    const int j = 2 * a + par;        // 0..63
    const float* gsrc = n + (size_t)(cb + col) * KDIM + 2 * j;
    float* ldst = buf + (a * 64 + 2 * col + 32 * par);
    *(v2f*)ldst = *(const v2f*)gsrc;
  }
#endif
}

// ---------------------------------------------------------------------------
// fused sim-GEMM + streaming sum-exp.
// Block: 8 waves x (16 rows each) sharing one 2048-col chunk; B tiles come
// from the shared LDS double buffer.  A fragments stay resident (64 VGPRs).
// fp32 WMMA layouts:
//   A 16x4 : lanes 0-15 (M=0..15) K={4k,4k+1}; lanes 16-31 K={4k+2,4k+3}
//   C      : VGPR m -> lanes 0-15 (M=m, N=lane); lanes 16-31 (M=m+8)
// ---------------------------------------------------------------------------
__global__ __launch_bounds__(256)
void sim_lse_kernel(const float* __restrict__ n,
                    float* __restrict__ pos,
                    float* __restrict__ psum) {
  __shared__ float buf[2][2048];      // 2 x 8KB tile buffers

  const int t      = threadIdx.x;
  const int waveid = t >> 5;
  const int lane   = t & 31;
  const int chunk  = blockIdx.x % NCHUNK;
  const int rowgrp = blockIdx.x / NCHUNK;

  const int rowbase = (rowgrp * WAVES_PER_BLOCK + waveid) * 16;
  const int posbase = (rowbase < BATCH) ? (rowbase + BATCH) : (rowbase - BATCH);
  const int hi = lane >> 4;
  const int l  = lane & 15;

  // resident A fragments: row = rowbase + l, K offset 2*hi
  const float* arow = n + (size_t)(rowbase + l) * KDIM + 2 * hi;
  v2f afrag[32];
  #pragma unroll
  for (int k = 0; k < 32; ++k)
    afrag[k] = *(const v2f*)(arow + 4 * k);

  float rs[8];
  #pragma unroll
  for (int m = 0; m < 8; ++m) rs[m] = 0.0f;

  const int colstart = chunk * CHUNK_COLS;

  // prime the double buffer (up to 2 stages = 8 async ops in flight)
  stage_tile(n, colstart, buf[0], t);
  stage_tile(n, colstart + 16, buf[1], t);

  for (int i = 0; i < NTILES; ++i) {
    const int cb = colstart + i * 16;

    // own stage(i) complete (async loads retire in order), then publish
    wait_async_le((i + 1 < NTILES) ? 4 : 0);
    __syncthreads();

    const float* bbuf = &buf[i & 1][2 * l + 32 * hi];
    v2f bfrag[32];
    #pragma unroll
    for (int k = 0; k < 32; ++k)
      bfrag[k] = *(const v2f*)(bbuf + 64 * k);   // conflict-free ds_load_b64

    // two independent accumulator chains
    v8f c0 = {0.f, 0.f, 0.f, 0.f, 0.f, 0.f, 0.f, 0.f};
    v8f c1 = {0.f, 0.f, 0.f, 0.f, 0.f, 0.f, 0.f, 0.f};
    #pragma unroll
    for (int k = 0; k < 32; k += 2) {
      c0 = __builtin_amdgcn_wmma_f32_16x16x4_f32(
               false, afrag[k],     false, bfrag[k],     (short)0, c0, false, false);
      c1 = __builtin_amdgcn_wmma_f32_16x16x4_f32(
               false, afrag[k + 1], false, bfrag[k + 1], (short)0, c1, false, false);
    }
    const v8f c = c0 + c1;

    if (cb == rowbase || cb == posbase) {
      // rare tiles (wave-uniform): diagonal mask / positive capture
      #pragma unroll
      for (int m = 0; m < 8; ++m) {
        const int slot = m + 8 * hi;
        const int row  = rowbase + slot;
        if (cb == posbase && l == slot) pos[row] = c[m];
        float e = __expf(c[m]);               // |c| <= 2 -> always safe
        if (cb == rowbase && l == slot) e = 0.0f;
        rs[m] += e;
      }
    } else {
      #pragma unroll
      for (int m = 0; m < 8; ++m)
        rs[m] += __expf(c[m]);
    }

    __syncthreads();                          // everyone done reading buf
    if (i + 2 < NTILES)
      stage_tile(n, cb + 32, buf[i & 1], t);  // overlap next DMA with WMMA
  }

  // single cross-lane reduction at the end (16-lane halves)
  #pragma unroll
  for (int m = 0; m < 8; ++m) {
    float s = rs[m];
    #pragma unroll
    for (int off = 8; off >= 1; off >>= 1) s += __shfl_xor(s, off, 16);
    if (l == m) {
      const int row = rowbase + m + 8 * hi;
      psum[row * NCHUNK + chunk] = s;
    }
  }
}

// ---------------------------------------------------------------------------
__global__ __launch_bounds__(256)
void lse_merge_kernel(const float* __restrict__ psum,
                      const float* __restrict__ pos,
                      float* __restrict__ acc) {
  const int row = blockIdx.x * blockDim.x + threadIdx.x;
  float contrib = 0.0f;
  if (row < TWO_B) {
    float S = 0.0f;
    #pragma unroll
    for (int c = 0; c < NCHUNK; ++c) S += psum[row * NCHUNK + c];
    contrib = logf(S) - pos[row];
  }
  contrib = wave_reduce_add(contrib);
  if ((threadIdx.x & 31) == 0) atomicAdd(&acc[0], contrib);
}

// ---------------------------------------------------------------------------
__global__ __launch_bounds__(256)
void sqdiff_kernel(const float* __restrict__ a, const float* __restrict__ b,
                   float* __restrict__ acc_slot, int n4) {
  float s = 0.0f;
  const int stride = gridDim.x * blockDim.x;
  for (int i = blockIdx.x * blockDim.x + threadIdx.x; i < n4; i += stride) {
    const float4 x = ((const float4*)a)[i];
    const float4 y = ((const float4*)b)[i];
    const float dx = x.x - y.x, dy = x.y - y.y, dz = x.z - y.z, dw = x.w - y.w;
    s += dx * dx + dy * dy + dz * dz + dw * dw;
  }
  s = wave_reduce_add(s);
  if ((threadIdx.x & 31) == 0) atomicAdd(acc_slot, s);
}

// ---------------------------------------------------------------------------
__global__ void finalize_kernel(const float* __restrict__ acc,
                                float* __restrict__ out) {
  if (threadIdx.x == 0) {
    const float closs  = acc[0] / (float)TWO_B;
    const float recon  = acc[1] / (float)RECON_N;
    const float zrecon = acc[2] / (float)ZREC_N;
    out[0] = recon + closs + zrecon;
    out[1] = closs;
    out[2] = recon;
    out[3] = zrecon;
  }
}

// ---------------------------------------------------------------------------
extern "C" void kernel_launch(void* const* d_in, const int* in_sizes, int n_in,
                              void* d_out, int out_size, void* d_ws, size_t ws_size,
                              hipStream_t stream) {
  (void)in_sizes; (void)n_in; (void)out_size; (void)ws_size;
  const float* rep    = (const float*)d_in[0];
  const float* xrecon = (const float*)d_in[1];
  const float* xorig  = (const float*)d_in[2];
  float* out = (float*)d_out;

  // ws layout (floats): n[8192*128] | pos[8192] | psum[8192*4] | acc[4]
  float* n    = (float*)d_ws;
  float* pos  = n    + (size_t)TWO_B * KDIM;
  float* psum = pos  + TWO_B;
  float* acc  = psum + (size_t)TWO_B * NCHUNK;

  zero_acc_kernel<<<1, 32, 0, stream>>>(acc);

  normalize_kernel<<<TWO_B / 8, 256, 0, stream>>>(rep, n);

  // (512 row-blocks / 8 per block) * 4 chunks = 256 blocks of 8 waves
  sim_lse_kernel<<<(TWO_B / 16 / WAVES_PER_BLOCK) * NCHUNK, 256, 0, stream>>>(
      n, pos, psum);

  lse_merge_kernel<<<TWO_B / 256, 256, 0, stream>>>(psum, pos, acc);

  sqdiff_kernel<<<1024, 256, 0, stream>>>(xrecon, xorig, &acc[1], RECON_N / 4);
  sqdiff_kernel<<<256, 256, 0, stream>>>(rep, rep + (size_t)BATCH * KDIM,
                                         &acc[2], ZREC_N / 4);

  finalize_kernel<<<1, 32, 0, stream>>>(acc, out);
}